// Reuse_Policy_2_83700322664899
// MI455X (gfx1250) — compile-verified
//
#include <hip/hip_runtime.h>
#include <math.h>

// ---------------- problem constants (from reference) ----------------
#define HH   512
#define WW   512
#define KK   3
#define CIN  3
#define C1N  20
#define C2N  40
#define H1   510
#define W1   510
#define H2   508
#define W2   508
#define NACT 18
#define KTOT ((size_t)C2N * H2 * W2)   // 10,322,560  (fc_w inner dim)

// gemv_wmma launch geometry (fixed -> per-wave chunk is compile-time)
#define GEMV_BLOCKS 1024
#define GEMV_TPB    256
#define GEMV_NWAVES (GEMV_BLOCKS * (GEMV_TPB / 32))                 // 8192
#define GEMV_PER    ((((long long)KTOT / 4 + GEMV_NWAVES - 1) / GEMV_NWAVES) * 4) // 1264 (79*16)

typedef __attribute__((ext_vector_type(2))) float v2f;
typedef __attribute__((ext_vector_type(8))) float v8f;

// ---------------- ws layout ----------------
// int   ws[0..3]   : bbox  mn_x, mx_x, mn_y, mx_y
// int   ws[4..15]  : reg[0..3]=mask1, reg[4..7]=rb (c1 write region), reg[8..11]=mask2
// float ws[32..49] : logits accumulator (18)
// float ws[64.. ]  : c1 blend buffer [20][510][510]  (~20.8 MB)

__global__ void init_kernel(int* __restrict__ bbox, float* __restrict__ logits) {
    if (threadIdx.x == 0) {
        bbox[0] = HH;   // min row
        bbox[1] = -1;   // max row
        bbox[2] = WW;   // min col
        bbox[3] = -1;   // max col
    }
    if (threadIdx.x < NACT) logits[threadIdx.x] = 0.f;
}

__global__ void bbox_kernel(const float* __restrict__ x,
                            const float* __restrict__ xp,
                            int* __restrict__ bbox) {
    int idx = blockIdx.x * blockDim.x + threadIdx.x;
    if (idx >= HH * WW) return;
    int h = idx / WW, w = idx % WW;
    bool diff = false;
    #pragma unroll
    for (int c = 0; c < CIN; ++c)
        diff |= (x[(size_t)c * HH * WW + idx] != xp[(size_t)c * HH * WW + idx]);
    if (diff) {
        atomicMin(&bbox[0], h);
        atomicMax(&bbox[1], h);
        atomicMin(&bbox[2], w);
        atomicMax(&bbox[3], w);
    }
}

__global__ void bounds_kernel(const int* __restrict__ bbox, int* __restrict__ reg) {
    if (threadIdx.x != 0 || blockIdx.x != 0) return;
    int mnx = bbox[0], mxx = bbox[1], mny = bbox[2], mxy = bbox[3];
    int r_x1 = max(mnx - (KK - 1), 0), r_x2 = min(mxx + KK, HH);
    int r_y1 = max(mny - (KK - 1), 0), r_y2 = min(mxy + KK, WW);
    // mask1 output region on the 510x510 conv1 grid
    reg[0] = r_x1; reg[1] = r_x2 - (KK - 1);
    reg[2] = r_y1; reg[3] = r_y2 - (KK - 1);
    // expanded region rb (c1 values conv2 needs -> blend/write region)
    int rb_x1 = max(r_x1 - (KK - 1), 0), rb_x2 = min(min(r_x2 + KK, HH), H1);
    int rb_y1 = max(r_y1 - (KK - 1), 0), rb_y2 = min(min(r_y2 + KK, WW), W1);
    reg[4] = rb_x1; reg[5] = rb_x2;
    reg[6] = rb_y1; reg[7] = rb_y2;
    // mask2 output region on the 508x508 conv2 grid
    reg[8]  = rb_x1; reg[9]  = rb_x2 - (KK - 1);
    reg[10] = rb_y1; reg[11] = rb_y2 - (KK - 1);
}

// blended c1 (new conv inside mask1, last_c1 elsewhere) over the rb region only
__global__ void c1_blend_kernel(const float* __restrict__ x,
                                const float* __restrict__ last_c1,
                                const float* __restrict__ w1,
                                const float* __restrict__ b1,
                                const int*   __restrict__ reg,
                                float*       __restrict__ c1buf) {
    __shared__ float sw[C1N * CIN * KK * KK];   // 540
    __shared__ float sb[C1N];
    for (int i = threadIdx.x; i < C1N * CIN * KK * KK; i += blockDim.x) sw[i] = w1[i];
    for (int i = threadIdx.x; i < C1N; i += blockDim.x) sb[i] = b1[i];
    __syncthreads();
    int pid = blockIdx.x * blockDim.x + threadIdx.x;
    if (pid >= H1 * W1) return;
    int h = pid / W1, w = pid % W1;
    if (h < reg[4] || h >= reg[5] || w < reg[6] || w >= reg[7]) return;
    bool inm1 = (h >= reg[0]) && (h < reg[1]) && (w >= reg[2]) && (w < reg[3]);
    if (inm1) {
        float xv[CIN][KK][KK];
        #pragma unroll
        for (int c = 0; c < CIN; ++c)
            #pragma unroll
            for (int i = 0; i < KK; ++i)
                #pragma unroll
                for (int j = 0; j < KK; ++j)
                    xv[c][i][j] = x[((size_t)c * HH + (h + i)) * WW + (w + j)];
        for (int oc = 0; oc < C1N; ++oc) {
            float acc = sb[oc];
            #pragma unroll
            for (int c = 0; c < CIN; ++c)
                #pragma unroll
                for (int i = 0; i < KK; ++i)
                    #pragma unroll
                    for (int j = 0; j < KK; ++j)
                        acc += xv[c][i][j] * sw[((oc * CIN + c) * KK + i) * KK + j];
            c1buf[((size_t)oc * H1 + h) * W1 + w] = fmaxf(acc, 0.f);
        }
    } else {
        for (int oc = 0; oc < C1N; ++oc)
            c1buf[((size_t)oc * H1 + h) * W1 + w] =
                last_c1[((size_t)oc * H1 + h) * W1 + w];
    }
}

// conv2 over mask2 region; accumulate fc_w[:,idx]*(relu(conv2)-last_c2[idx])
__global__ void conv2_delta_kernel(const float* __restrict__ c1buf,
                                   const float* __restrict__ last_c2,
                                   const float* __restrict__ w2,
                                   const float* __restrict__ b2,
                                   const float* __restrict__ fcw,
                                   const int*   __restrict__ reg,
                                   float*       __restrict__ logits) {
    __shared__ float sw[C2N * C1N * KK * KK];   // 7200 floats = 28.8 KB
    __shared__ float sb[C2N];
    __shared__ float red[NACT];
    for (int i = threadIdx.x; i < C2N * C1N * KK * KK; i += blockDim.x) sw[i] = w2[i];
    for (int i = threadIdx.x; i < C2N; i += blockDim.x) sb[i] = b2[i];
    if (threadIdx.x < NACT) red[threadIdx.x] = 0.f;
    __syncthreads();

    int pid = blockIdx.x * blockDim.x + threadIdx.x;
    bool active = false;
    int h = 0, w = 0;
    if (pid < H2 * W2) {
        h = pid / W2; w = pid % W2;
        active = (h >= reg[8]) && (h < reg[9]) && (w >= reg[10]) && (w < reg[11]);
    }
    if (active) {
        float acc[C2N];
        #pragma unroll
        for (int oc = 0; oc < C2N; ++oc) acc[oc] = sb[oc];
        for (int ic = 0; ic < C1N; ++ic)
            #pragma unroll
            for (int i = 0; i < KK; ++i)
                #pragma unroll
                for (int j = 0; j < KK; ++j) {
                    float v = c1buf[((size_t)ic * H1 + (h + i)) * W1 + (w + j)];
                    #pragma unroll
                    for (int oc = 0; oc < C2N; ++oc)
                        acc[oc] += v * sw[((oc * C1N + ic) * KK + i) * KK + j];
                }
        float part[NACT];
        #pragma unroll
        for (int a = 0; a < NACT; ++a) part[a] = 0.f;
        for (int oc = 0; oc < C2N; ++oc) {
            size_t idx = ((size_t)oc * H2 + h) * W2 + w;
            float delta = fmaxf(acc[oc], 0.f) - last_c2[idx];
            #pragma unroll
            for (int a = 0; a < NACT; ++a)
                part[a] += fcw[(size_t)a * KTOT + idx] * delta;
        }
        #pragma unroll
        for (int a = 0; a < NACT; ++a) atomicAdd(&red[a], part[a]);
    }
    __syncthreads();
    if (threadIdx.x < NACT && red[threadIdx.x] != 0.f)
        atomicAdd(&logits[threadIdx.x], red[threadIdx.x]);
}

// -------- dominant kernel: logits[0..15] += fc_w[0:16,:] @ last_c2 via WMMA --------
// A tile = fc_w rows 0..15, K-chunk of 4 (lane L: M=L&15, K-pair=(L>>4)*2).
// B tile = the same 4 v-values broadcast to all 16 columns (VGPR v, half h -> K=v+2h).
// All 16 D-columns are identical; lanes 0 and 16 (N==0) hold M=0..7 / 8..15.
// Per-wave chunk GEMV_PER is compile-time (no runtime division) and a multiple
// of 16, as is KTOT, so the 4-deep unrolled body covers every wave exactly.
__global__ void __launch_bounds__(GEMV_TPB)
gemv_wmma_kernel(const float* __restrict__ fcw,
                 const float* __restrict__ v,
                 float*       __restrict__ logits) {
    const long long Ktot = (long long)KTOT;
    int wave = blockIdx.x * (GEMV_TPB >> 5) + (threadIdx.x >> 5);
    int lane = threadIdx.x & 31;

    long long k0 = (long long)wave * GEMV_PER;
    long long ke = k0 + GEMV_PER; if (ke > Ktot) ke = Ktot;

    const int m    = lane & 15;
    const int half = lane >> 4;               // 0 or 1 -> K-pair offset
    // 8-byte aligned float2 streams (k multiple of 4, koff even, rows even)
    const v2f* __restrict__ arow2 = (const v2f*)(fcw + (size_t)m * KTOT);
    const v2f* __restrict__ vv2   = (const v2f*)v;

    v8f c0 = {}, c1 = {}, c2 = {}, c3 = {};
    long long k = k0;
    for (; k + 16 <= ke; k += 16) {
        long long p = (k >> 1) + half;        // float2 index of this lane's pair
        __builtin_prefetch((const float*)(arow2 + p) + 1024, 0, 0);
        v2f a0 = arow2[p],     b0 = vv2[p];
        v2f a1 = arow2[p + 2], b1 = vv2[p + 2];
        v2f a2 = arow2[p + 4], b2 = vv2[p + 4];
        v2f a3 = arow2[p + 6], b3 = vv2[p + 6];
        c0 = __builtin_amdgcn_wmma_f32_16x16x4_f32(false, a0, false, b0,
                                                   (short)0, c0, false, false);
        c1 = __builtin_amdgcn_wmma_f32_16x16x4_f32(false, a1, false, b1,
                                                   (short)0, c1, false, false);
        c2 = __builtin_amdgcn_wmma_f32_16x16x4_f32(false, a2, false, b2,
                                                   (short)0, c2, false, false);
        c3 = __builtin_amdgcn_wmma_f32_16x16x4_f32(false, a3, false, b3,
                                                   (short)0, c3, false, false);
    }
    for (; k < ke; k += 4) {                  // safety remainder (never taken)
        long long p = (k >> 1) + half;
        v2f a0 = arow2[p], b0 = vv2[p];
        c0 = __builtin_amdgcn_wmma_f32_16x16x4_f32(false, a0, false, b0,
                                                   (short)0, c0, false, false);
    }
    if ((lane & 15) == 0) {                   // N==0 column: lanes 0 and 16
        int mbase = half * 8;                 // lane0 -> M 0..7, lane16 -> M 8..15
        #pragma unroll
        for (int i = 0; i < 8; ++i)
            atomicAdd(&logits[mbase + i], (c0[i] + c1[i]) + (c2[i] + c3[i]));
    }
}

// rows 16..17 of fc_w (18 is not a multiple of 16): plain vectorized GEMV tail
__global__ void gemv_tail_kernel(const float* __restrict__ fcw,
                                 const float* __restrict__ v,
                                 float*       __restrict__ logits) {
    const long long K4 = (long long)(KTOT / 4);
    const float4* __restrict__ r16 = (const float4*)(fcw + (size_t)16 * KTOT);
    const float4* __restrict__ r17 = (const float4*)(fcw + (size_t)17 * KTOT);
    const float4* __restrict__ vv  = (const float4*)v;
    long long i0     = (long long)blockIdx.x * blockDim.x + threadIdx.x;
    long long stride = (long long)gridDim.x * blockDim.x;
    float a16 = 0.f, a17 = 0.f;
    for (long long i = i0; i < K4; i += stride) {
        float4 x = vv[i];
        float4 w6 = r16[i];
        float4 w7 = r17[i];
        a16 += x.x * w6.x + x.y * w6.y + x.z * w6.z + x.w * w6.w;
        a17 += x.x * w7.x + x.y * w7.y + x.z * w7.z + x.w * w7.w;
    }
    #pragma unroll
    for (int off = 16; off > 0; off >>= 1) {
        a16 += __shfl_down(a16, off, 32);
        a17 += __shfl_down(a17, off, 32);
    }
    if ((threadIdx.x & 31) == 0) {
        atomicAdd(&logits[16], a16);
        atomicAdd(&logits[17], a17);
    }
}

__global__ void softmax_kernel(const float* __restrict__ logits, float* __restrict__ out) {
    if (threadIdx.x != 0 || blockIdx.x != 0) return;
    float mx = -INFINITY;
    for (int a = 0; a < NACT; ++a) mx = fmaxf(mx, logits[a]);
    float e[NACT]; float s = 0.f;
    for (int a = 0; a < NACT; ++a) { e[a] = expf(logits[a] - mx); s += e[a]; }
    float inv = 1.f / s;
    for (int a = 0; a < NACT; ++a) out[a] = e[a] * inv;
}

extern "C" void kernel_launch(void* const* d_in, const int* in_sizes, int n_in,
                              void* d_out, int out_size, void* d_ws, size_t ws_size,
                              hipStream_t stream) {
    const float* x       = (const float*)d_in[0];
    const float* x_prev  = (const float*)d_in[1];
    const float* last_c1 = (const float*)d_in[2];
    const float* last_c2 = (const float*)d_in[3];
    const float* conv1_w = (const float*)d_in[4];
    const float* conv1_b = (const float*)d_in[5];
    const float* conv2_w = (const float*)d_in[6];
    const float* conv2_b = (const float*)d_in[7];
    const float* fc_w    = (const float*)d_in[8];
    float* out = (float*)d_out;

    int*   bbox   = (int*)d_ws;
    int*   reg    = bbox + 4;
    float* logits = (float*)d_ws + 32;
    float* c1buf  = (float*)d_ws + 64;   // 20*510*510 floats (~20.8 MB)

    init_kernel<<<1, 32, 0, stream>>>(bbox, logits);
    bbox_kernel<<<(HH * WW + 255) / 256, 256, 0, stream>>>(x, x_prev, bbox);
    bounds_kernel<<<1, 1, 0, stream>>>(bbox, reg);
    c1_blend_kernel<<<(H1 * W1 + 255) / 256, 256, 0, stream>>>(
        x, last_c1, conv1_w, conv1_b, reg, c1buf);
    conv2_delta_kernel<<<(H2 * W2 + 255) / 256, 256, 0, stream>>>(
        c1buf, last_c2, conv2_w, conv2_b, fc_w, reg, logits);
    gemv_wmma_kernel<<<GEMV_BLOCKS, GEMV_TPB, 0, stream>>>(fc_w, last_c2, logits);
    gemv_tail_kernel<<<512, 256, 0, stream>>>(fc_w, last_c2, logits);
    softmax_kernel<<<1, 32, 0, stream>>>(logits, out);
}